// GCN_64836826300768
// MI455X (gfx1250) — compile-verified
//
#include <hip/hip_runtime.h>
#include <math.h>

// Problem constants (match reference)
#define FIN 128
#define HIDF 64
#define NCLS 16

typedef __attribute__((ext_vector_type(2))) float v2f;
typedef __attribute__((ext_vector_type(8))) float v8f;

// ---------------- degree / norm ----------------
__global__ void deg_init_kernel(float* __restrict__ deg, int n) {
  int i = blockIdx.x * blockDim.x + threadIdx.x;
  if (i < n) deg[i] = 1.0f; // self-loop contributes 1 to in-degree
}

__global__ void deg_count_kernel(const long long* __restrict__ col,
                                 float* __restrict__ deg, int ne) {
  int i = blockIdx.x * blockDim.x + threadIdx.x;
  if (i < ne) atomicAdd(&deg[(int)col[i]], 1.0f);
}

__global__ void rsqrt_inplace_kernel(float* __restrict__ d, int n) {
  int i = blockIdx.x * blockDim.x + threadIdx.x;
  if (i < n) d[i] = rsqrtf(d[i]); // deg >= 1 always (self-loop)
}

// ---------------- fp32 WMMA GEMM: C[M,Nc] = (RELU?max(A,0):A)[M,K] * B[K,Nc] ----------------
// One wave per 16x16 output tile; K stepped by 4 via V_WMMA_F32_16X16X4_F32.
template <int RELU>
__global__ void wmma_gemm_f32(const float* __restrict__ A, const float* __restrict__ B,
                              float* __restrict__ C, int M, int K, int Nc) {
  const int lane = threadIdx.x & 31;
  const int wave = (int)((blockIdx.x * blockDim.x + threadIdx.x) >> 5);
  const int tilesN = Nc >> 4;
  const int tm = wave / tilesN;
  const int tn = wave % tilesN;
  if (tm * 16 >= M) return;           // wave-uniform: EXEC stays all-1s for WMMA
  const int hl = lane >> 4;           // 0: lanes 0-15, 1: lanes 16-31
  const int lrow = lane & 15;

  const float* arow = A + (size_t)(tm * 16 + lrow) * K;
  const float* bcol = B + (size_t)tn * 16 + lrow;

  v8f acc = {};
  for (int k0 = 0; k0 < K; k0 += 4) {
    // A 16x4: lanes 0-15 hold K={0,1}, lanes 16-31 hold K={2,3}
    float a0 = arow[k0 + 2 * hl + 0];
    float a1 = arow[k0 + 2 * hl + 1];
    if (RELU) { a0 = fmaxf(a0, 0.0f); a1 = fmaxf(a1, 0.0f); }
    v2f a; a.x = a0; a.y = a1;
    // B 4x16: V0 = rows K={0 (lanes 0-15), 1 (lanes 16-31)}, V1 = rows K={2,3}
    v2f b;
    b.x = bcol[(size_t)(k0 + hl) * Nc];
    b.y = bcol[(size_t)(k0 + 2 + hl) * Nc];
    acc = __builtin_amdgcn_wmma_f32_16x16x4_f32(
        /*neg_a=*/false, a, /*neg_b=*/false, b,
        /*c_mod=*/(short)0, acc, /*reuse_a=*/false, /*reuse_b=*/false);
  }
  float* crow = C + (size_t)(tm * 16 + hl * 8) * Nc + (size_t)tn * 16 + lrow;
#pragma unroll
  for (int r = 0; r < 8; ++r) crow[(size_t)r * Nc] = acc[r];
}

// ---------------- out = bias + h * dinv^2  (self-loop term fused with bias init), float4 ----------------
template <int F>
__global__ void init_out4_kernel(const float* __restrict__ h, const float* __restrict__ dinv,
                                 const float* __restrict__ bias, float* __restrict__ out,
                                 int n) {
  constexpr int FQ = F / 4;
  long long idx = (long long)blockIdx.x * blockDim.x + threadIdx.x;
  if (idx >= (long long)n * FQ) return;
  int i = (int)(idx / FQ);
  int q = (int)(idx % FQ);
  float di = dinv[i];
  float d2 = di * di;
  float4 hv = ((const float4*)h)[idx];
  float4 bv = ((const float4*)bias)[q];
  float4 o;
  o.x = bv.x + hv.x * d2;
  o.y = bv.y + hv.y * d2;
  o.z = bv.z + hv.z * d2;
  o.w = bv.w + hv.w * d2;
  ((float4*)out)[idx] = o;
}

// ---------------- edge scatter: out[col] += h[row] * dinv[row]*dinv[col] ----------------
// One thread per (edge, float4 chunk): b128 gather + 4 float atomics.
// h/out node tensors (6.4-25.6 MB) are resident in the 192 MB L2, so the random
// gather and the atomic RMWs are L2-side traffic, not HBM.
template <int F>
__global__ void scatter4_kernel(const long long* __restrict__ row, const long long* __restrict__ col,
                                const float* __restrict__ dinv, const float* __restrict__ h,
                                float* __restrict__ out, int ne) {
  constexpr int FQ = F / 4;
  long long idx = (long long)blockIdx.x * blockDim.x + threadIdx.x;
  if (idx >= (long long)ne * FQ) return;
  int e = (int)(idx / FQ);
  int q = (int)(idx % FQ);
  int r = (int)row[e];
  int c = (int)col[e];
  float nrm = dinv[r] * dinv[c];
  float4 hv = ((const float4*)(h + (long long)r * F))[q];
  float* o = out + (long long)c * F + (long long)q * 4;
  atomicAdd(o + 0, hv.x * nrm);
  atomicAdd(o + 1, hv.y * nrm);
  atomicAdd(o + 2, hv.z * nrm);
  atomicAdd(o + 3, hv.w * nrm);
}

// ---------------- row-wise log_softmax over 16 classes (vectorized) ----------------
__global__ void log_softmax16_kernel(const float* __restrict__ in, float* __restrict__ out, int n) {
  int i = blockIdx.x * blockDim.x + threadIdx.x;
  if (i >= n) return;
  const float4* r = (const float4*)(in + (size_t)i * NCLS);
  float4 v[4];
#pragma unroll
  for (int j = 0; j < 4; ++j) v[j] = r[j];
  float m = v[0].x;
#pragma unroll
  for (int j = 0; j < 4; ++j) {
    m = fmaxf(m, v[j].x); m = fmaxf(m, v[j].y);
    m = fmaxf(m, v[j].z); m = fmaxf(m, v[j].w);
  }
  float s = 0.0f;
#pragma unroll
  for (int j = 0; j < 4; ++j) {
    s += __expf(v[j].x - m); s += __expf(v[j].y - m);
    s += __expf(v[j].z - m); s += __expf(v[j].w - m);
  }
  float off = m + __logf(s);
  float4* o = (float4*)(out + (size_t)i * NCLS);
#pragma unroll
  for (int j = 0; j < 4; ++j) {
    float4 w;
    w.x = v[j].x - off; w.y = v[j].y - off;
    w.z = v[j].z - off; w.w = v[j].w - off;
    o[j] = w;
  }
}

extern "C" void kernel_launch(void* const* d_in, const int* in_sizes, int n_in,
                              void* d_out, int out_size, void* d_ws, size_t ws_size,
                              hipStream_t stream) {
  const float* x       = (const float*)d_in[0];
  const long long* ei  = (const long long*)d_in[1]; // int64 edge_index [2, E]
  const float* W1      = (const float*)d_in[2];
  const float* b1      = (const float*)d_in[3];
  const float* W2      = (const float*)d_in[4];
  const float* b2      = (const float*)d_in[5];

  const int n  = in_sizes[0] / FIN;   // 100000
  const int ne = in_sizes[1] / 2;     // 1600000
  const long long* row = ei;
  const long long* col = ei + ne;

  // workspace layout (out2 aliases h1, which is dead after scatter-1)
  char* ws = (char*)d_ws;
  size_t off = 0;
  auto carve = [&](size_t bytes) -> void* {
    void* p = ws + off;
    off += (bytes + 255) & ~(size_t)255;
    return p;
  };
  float* dinv = (float*)carve((size_t)n * sizeof(float));
  float* h1   = (float*)carve((size_t)n * HIDF * sizeof(float));
  float* out1 = (float*)carve((size_t)n * HIDF * sizeof(float));
  float* h2   = (float*)carve((size_t)n * NCLS * sizeof(float));
  float* out2 = h1; // alias: h1 no longer needed once scatter-1 has run

  const int B = 256;
  // 1) normalization coefficients
  deg_init_kernel<<<(n + B - 1) / B, B, 0, stream>>>(dinv, n);
  deg_count_kernel<<<(ne + B - 1) / B, B, 0, stream>>>(col, dinv, ne);
  rsqrt_inplace_kernel<<<(n + B - 1) / B, B, 0, stream>>>(dinv, n);

  // 2) layer 1: h1 = x @ W1   (fp32 WMMA, one wave per 16x16 tile)
  {
    long long thr = (long long)(n / 16) * (HIDF / 16) * 32;
    wmma_gemm_f32<0><<<(int)((thr + B - 1) / B), B, 0, stream>>>(x, W1, h1, n, FIN, HIDF);
  }
  // out1 = b1 + h1*dinv^2 ; then scatter E edges (float4 chunks)
  init_out4_kernel<HIDF><<<(int)(((long long)n * (HIDF / 4) + B - 1) / B), B, 0, stream>>>(h1, dinv, b1, out1, n);
  scatter4_kernel<HIDF><<<(int)(((long long)ne * (HIDF / 4) + B - 1) / B), B, 0, stream>>>(row, col, dinv, h1, out1, ne);

  // 3) layer 2: h2 = relu(out1) @ W2  (ReLU fused into A-load)
  {
    long long thr = (long long)((n + 15) / 16) * (NCLS / 16) * 32;
    wmma_gemm_f32<1><<<(int)((thr + B - 1) / B), B, 0, stream>>>(out1, W2, h2, n, HIDF, NCLS);
  }
  init_out4_kernel<NCLS><<<(int)(((long long)n * (NCLS / 4) + B - 1) / B), B, 0, stream>>>(h2, dinv, b2, out2, n);
  scatter4_kernel<NCLS><<<(int)(((long long)ne * (NCLS / 4) + B - 1) / B), B, 0, stream>>>(row, col, dinv, h2, out2, ne);

  // 4) log_softmax rows of 16 -> d_out
  log_softmax16_kernel<<<(n + B - 1) / B, B, 0, stream>>>(out2, (float*)d_out, n);
}